// CustomLoss_24644522344572
// MI455X (gfx1250) — compile-verified
//
#include <hip/hip_runtime.h>
#include <hip/hip_bf16.h>

// ---------------------------------------------------------------------------
// MultiBox-style loss with hard-negative mining on gfx1250 (MI455X).
// Bandwidth-bound: 168MB input read once (K1); 8MB key array read twice
// (K3,K5). Top-k selection via 2-level 12-bit radix histograms on a
// monotone u32 mapping of logp0. Wave reductions use V_WMMA_F32_16X16X4_F32.
// ---------------------------------------------------------------------------

#define NC    6
#define ROW   10
#define A_    131072
#define B_    16
#define NBLK  ((B_ * A_) / 256)      // 8192 blocks of 256 threads, 1 anchor/thread
#define NBINS 4096

typedef __attribute__((ext_vector_type(8))) float v8f;
typedef __attribute__((ext_vector_type(2))) float v2f;

struct Ctl {
    float num_pos_f, pos_loss, box;
    int   num_pos, num_neg, T1;
    unsigned rem1;
    int   T2;
    unsigned rem2, nT2;
    int   pad0, pad1;                 // 48 bytes
};

// Workspace layout (bytes)
static constexpr size_t OFF_KEYS = 0;                                  // u32[B_*A_]  = 8 MiB
static constexpr size_t OFF_HIST = (size_t)B_ * A_ * 4;                // hist1+hist2: 2*16*4096 u32
static constexpr size_t OFF_P1   = OFF_HIST + (size_t)2 * B_ * NBINS * 4; // f32[NBLK][3]
static constexpr size_t OFF_P2   = OFF_P1 + (size_t)NBLK * 3 * 4;      // f32[NBLK][2]
static constexpr size_t OFF_CTL  = OFF_P2 + (size_t)NBLK * 2 * 4;      // Ctl[16]

// ---------------------------------------------------------------------------
// Exact f32 32-lane sum using V_WMMA_F32_16X16X4_F32.
// A(16x4): lane L<16 holds A[L][K=0..1] in vgpr0..1, lane L>=16 holds A[L-16][K=2..3].
// With A = [x, 0] and B = ones: D[m][n] = x[m] + x[m+16] for every n.
// C/D layout: lane L<16, vgpr r -> D[r][L]; lane L>=16, vgpr r -> D[8+r][L-16].
// So sum of a lane's 8 result vgprs = half-sum; xor-16 shuffle finishes it.
// ---------------------------------------------------------------------------
__device__ __forceinline__ float wave_sum32(float x) {
    v2f a; a.x = x;    a.y = 0.0f;
    v2f o; o.x = 1.0f; o.y = 1.0f;
    v8f c = {0.f, 0.f, 0.f, 0.f, 0.f, 0.f, 0.f, 0.f};
    v8f d = __builtin_amdgcn_wmma_f32_16x16x4_f32(false, a, false, o,
                                                  (short)0, c, false, false);
    float s = ((d[0] + d[1]) + (d[2] + d[3])) + ((d[4] + d[5]) + (d[6] + d[7]));
    s += __shfl_xor(s, 16);
    return s;   // full 32-lane sum, broadcast to all lanes
}

// ---------------------------------------------------------------------------
// K0: zero hist1 + hist2
// ---------------------------------------------------------------------------
__global__ __launch_bounds__(256) void k0_zero(unsigned* __restrict__ hist) {
    int i = blockIdx.x * 256 + threadIdx.x;
    if (i < 2 * B_ * NBINS) hist[i] = 0u;
}

// ---------------------------------------------------------------------------
// K1: main streaming pass. Per anchor: CE terms, key(logp0), smooth-L1 box.
// Emits: keys[], integer hist1 (negatives only), per-block float partials.
// ---------------------------------------------------------------------------
__global__ __launch_bounds__(256) void k1_main(const float* __restrict__ y_pre,
                                               const float* __restrict__ y_batch,
                                               unsigned* __restrict__ keys,
                                               unsigned* __restrict__ hist1,
                                               float* __restrict__ part1) {
    const int blk = blockIdx.x;
    const int b   = blk >> 9;                       // 512 blocks per sample
    const int idx = blk * 256 + threadIdx.x;        // global anchor id = b*A_ + a
    const float* __restrict__ p = y_pre   + idx * ROW;
    const float* __restrict__ q = y_batch + idx * ROW;

    float cp[NC], bp[4], ch[NC], bh[4];
#pragma unroll
    for (int i = 0; i < NC; ++i) cp[i] = p[i];
#pragma unroll
    for (int i = 0; i < 4; ++i)  bp[i] = p[NC + i];
#pragma unroll
    for (int i = 0; i < NC; ++i) ch[i] = q[i];
#pragma unroll
    for (int i = 0; i < 4; ++i)  bh[i] = q[NC + i];

    // pos mask + label (first-occurrence argmax, matches jnp.argmax)
    float mx = ch[0]; int lab = 0;
#pragma unroll
    for (int i = 1; i < NC; ++i) { if (ch[i] > mx) { mx = ch[i]; lab = i; } }
    const bool pos = mx > 0.0f;

    // log-softmax pieces
    float m = cp[0];
#pragma unroll
    for (int i = 1; i < NC; ++i) m = fmaxf(m, cp[i]);
    float s = 0.0f;
#pragma unroll
    for (int i = 0; i < NC; ++i) s += __expf(cp[i] - m);
    const float lse = m + __logf(s);

    float cplab = cp[0];
#pragma unroll
    for (int i = 1; i < NC; ++i) cplab = (lab == i) ? cp[i] : cplab;
    const float ce    = lse - cplab;     // -logp[label]
    const float logp0 = cp[0] - lse;

    // monotone float -> u32 key (ascending logp0 == ascending key)
    const unsigned u   = __float_as_uint(logp0);
    const unsigned key = (u & 0x80000000u) ? ~u : (u | 0x80000000u);

    // smooth-L1 box term
    float box = 0.0f;
#pragma unroll
    for (int i = 0; i < 4; ++i) {
        float d = fabsf(bp[i] - bh[i]);
        box += (d < 1.0f) ? 0.5f * d * d : d - 0.5f;
    }

    keys[idx] = pos ? 0xFFFFFFFFu : key;            // sentinel excludes positives
    if (!pos) atomicAdd(&hist1[(b << 12) + (key >> 20)], 1u);

    // deterministic per-block reduction (WMMA wave sums -> LDS -> thread 0)
    const float sp = wave_sum32(pos ? 1.0f : 0.0f);
    const float sl = wave_sum32(pos ? ce   : 0.0f);
    const float sb = wave_sum32(pos ? box  : 0.0f);

    __shared__ float acc[8][3];
    const int wave = threadIdx.x >> 5, lane = threadIdx.x & 31;
    if (lane == 0) { acc[wave][0] = sp; acc[wave][1] = sl; acc[wave][2] = sb; }
    __syncthreads();
    if (threadIdx.x == 0) {
        float a0 = 0.f, a1 = 0.f, a2 = 0.f;
#pragma unroll
        for (int w = 0; w < 8; ++w) { a0 += acc[w][0]; a1 += acc[w][1]; a2 += acc[w][2]; }
        part1[blk * 3 + 0] = a0;
        part1[blk * 3 + 1] = a1;
        part1[blk * 3 + 2] = a2;
    }
}

// ---------------------------------------------------------------------------
// K2: per-sample: reduce 512 partials (fixed order), compute num_neg,
//     scan hist1 for level-1 threshold bin T1.
// ---------------------------------------------------------------------------
__global__ __launch_bounds__(256) void k2_sel1(const float* __restrict__ part1,
                                               const unsigned* __restrict__ hist1,
                                               Ctl* __restrict__ ctl) {
    const int b = blockIdx.x, tid = threadIdx.x;
    __shared__ float r0[256], r1[256], r2[256];
    float a0 = 0.f, a1 = 0.f, a2 = 0.f;
    for (int i = tid; i < 512; i += 256) {
        const float* pp = part1 + (b * 512 + i) * 3;
        a0 += pp[0]; a1 += pp[1]; a2 += pp[2];
    }
    r0[tid] = a0; r1[tid] = a1; r2[tid] = a2;
    __syncthreads();
    for (int st = 128; st > 0; st >>= 1) {
        if (tid < st) { r0[tid] += r0[tid + st]; r1[tid] += r1[tid + st]; r2[tid] += r2[tid + st]; }
        __syncthreads();
    }
    __shared__ unsigned hb[NBINS];
    for (int i = tid; i < NBINS; i += 256) hb[i] = hist1[(b << 12) + i];
    __syncthreads();

    if (tid == 0) {
        Ctl c;
        c.num_pos_f = r0[0]; c.pos_loss = r1[0]; c.box = r2[0];
        const int num_pos = (int)(r0[0] + 0.5f);
        const int nna     = A_ - num_pos;
        const int num_neg = (3 * num_pos < nna) ? 3 * num_pos : nna;
        c.num_pos = num_pos; c.num_neg = num_neg;
        c.T1 = -1; c.rem1 = 0; c.T2 = -1; c.rem2 = 0; c.nT2 = 0;
        c.pad0 = 0; c.pad1 = 0;
        if (num_neg > 0) {
            unsigned cum = 0;
            for (int i = 0; i < NBINS; ++i) {
                unsigned h = hb[i];
                if (cum + h >= (unsigned)num_neg) { c.T1 = i; c.rem1 = (unsigned)num_neg - cum; break; }
                cum += h;
            }
        }
        ctl[b] = c;
    }
}

// ---------------------------------------------------------------------------
// K3: level-2 histogram of keys that land in bin T1 (bits 19:8).
// ---------------------------------------------------------------------------
__global__ __launch_bounds__(256) void k3_hist2(const unsigned* __restrict__ keys,
                                                const Ctl* __restrict__ ctl,
                                                unsigned* __restrict__ hist2) {
    const int blk = blockIdx.x;
    const int b   = blk >> 9;
    const int T1  = ctl[b].T1;
    if (T1 < 0) return;
    const unsigned key = keys[blk * 256 + threadIdx.x];
    if ((int)(key >> 20) == T1)
        atomicAdd(&hist2[(b << 12) + ((key >> 8) & 0xFFFu)], 1u);
}

// ---------------------------------------------------------------------------
// K4: per-sample scan of hist2 -> T2, rem2, nT2.
// ---------------------------------------------------------------------------
__global__ __launch_bounds__(256) void k4_sel2(const unsigned* __restrict__ hist2,
                                               Ctl* __restrict__ ctl) {
    const int b = blockIdx.x, tid = threadIdx.x;
    __shared__ unsigned hb[NBINS];
    for (int i = tid; i < NBINS; i += 256) hb[i] = hist2[(b << 12) + i];
    __syncthreads();
    if (tid == 0) {
        Ctl c = ctl[b];
        if (c.T1 >= 0) {
            unsigned cum = 0, target = c.rem1;
            for (int i = 0; i < NBINS; ++i) {
                unsigned h = hb[i];
                if (cum + h >= target) { c.T2 = i; c.rem2 = target - cum; c.nT2 = h; break; }
                cum += h;
            }
        }
        ctl[b] = c;
    }
}

// ---------------------------------------------------------------------------
// K5: sum ce0 of selected negatives: everything strictly below the threshold
//     24-bit prefix, plus the boundary sub-bin separately (for proportional
//     tie resolution). Deterministic WMMA/block reduction -> fixed partials.
// ---------------------------------------------------------------------------
__global__ __launch_bounds__(256) void k5_sum(const unsigned* __restrict__ keys,
                                              const Ctl* __restrict__ ctl,
                                              float* __restrict__ part2) {
    const int blk = blockIdx.x;
    const int b   = blk >> 9;
    const int T1  = ctl[b].T1;
    const int T2  = ctl[b].T2;
    const unsigned key = keys[blk * 256 + threadIdx.x];

    float sBel = 0.0f, sT2 = 0.0f;
    if (T1 >= 0) {
        const int bin = (int)(key >> 20);          // sentinel 0xFFF.. -> bin 4095 > T1
        if (bin <= T1) {
            const unsigned ub = (key & 0x80000000u) ? (key & 0x7FFFFFFFu) : ~key;
            const float ce0 = -__uint_as_float(ub);  // -logp0
            if (bin < T1) {
                sBel = ce0;
            } else {
                const int mid = (int)((key >> 8) & 0xFFFu);
                if (mid < T2)       sBel = ce0;
                else if (mid == T2) sT2 = ce0;
            }
        }
    }
    const float rb = wave_sum32(sBel);
    const float rt = wave_sum32(sT2);

    __shared__ float acc[8][2];
    const int wave = threadIdx.x >> 5, lane = threadIdx.x & 31;
    if (lane == 0) { acc[wave][0] = rb; acc[wave][1] = rt; }
    __syncthreads();
    if (threadIdx.x == 0) {
        float a0 = 0.f, a1 = 0.f;
#pragma unroll
        for (int w = 0; w < 8; ++w) { a0 += acc[w][0]; a1 += acc[w][1]; }
        part2[blk * 2 + 0] = a0;
        part2[blk * 2 + 1] = a1;
    }
}

// ---------------------------------------------------------------------------
// K6: finalize. Reduce K5 partials per sample (fixed order), apply reference
//     edge-case logic, write (total_loss, L_class, L_box).
// ---------------------------------------------------------------------------
__global__ __launch_bounds__(256) void k6_final(const float* __restrict__ part2,
                                                const Ctl* __restrict__ ctl,
                                                float* __restrict__ out) {
    const int tid = threadIdx.x;
    __shared__ float r0[256], r1[256];
    __shared__ float negB[B_], negT[B_];

    for (int b = 0; b < B_; ++b) {
        float a0 = 0.f, a1 = 0.f;
        for (int i = tid; i < 512; i += 256) {
            const float* pp = part2 + (b * 512 + i) * 2;
            a0 += pp[0]; a1 += pp[1];
        }
        r0[tid] = a0; r1[tid] = a1;
        __syncthreads();
        for (int st = 128; st > 0; st >>= 1) {
            if (tid < st) { r0[tid] += r0[tid + st]; r1[tid] += r1[tid + st]; }
            __syncthreads();
        }
        if (tid == 0) { negB[b] = r0[0]; negT[b] = r1[0]; }
        __syncthreads();
    }

    if (tid == 0) {
        float Lc_sum = 0.f, total_pos = 0.f, box_sum = 0.f;
        int n_valid = 0;
        for (int b = 0; b < B_; ++b) {
            Ctl c = ctl[b];
            float frac = 0.f;
            if (c.nT2 > 0 && c.rem2 > 0)
                frac = negT[b] * ((float)c.rem2 / (float)c.nT2);
            const float neg_loss = negB[b] + frac;
            const bool has_pos = c.num_pos > 0;
            const bool has_neg = (A_ - c.num_pos) > 0;
            const float denom = has_neg ? (float)(c.num_neg + c.num_pos) : (float)c.num_pos;
            float ps = (has_neg ? c.pos_loss + neg_loss : c.pos_loss) / fmaxf(denom, 1.0f);
            ps = has_pos ? ps : 0.0f;
            Lc_sum   += ps;
            n_valid  += has_pos ? 1 : 0;
            total_pos += (float)c.num_pos;
            box_sum  += c.box;
        }
        const float L_class = (n_valid > 0) ? Lc_sum / (float)n_valid : 0.0f;
        const float L_box   = (total_pos > 0.f) ? box_sum / (total_pos + 1e-6f) : 0.0f;
        out[0] = L_class + 0.5f * L_box;
        out[1] = L_class;
        out[2] = L_box;
    }
}

// ---------------------------------------------------------------------------
extern "C" void kernel_launch(void* const* d_in, const int* in_sizes, int n_in,
                              void* d_out, int out_size, void* d_ws, size_t ws_size,
                              hipStream_t stream) {
    const float* y_pre   = (const float*)d_in[0];
    const float* y_batch = (const float*)d_in[1];
    char* ws = (char*)d_ws;

    unsigned* keys  = (unsigned*)(ws + OFF_KEYS);
    unsigned* hist1 = (unsigned*)(ws + OFF_HIST);
    unsigned* hist2 = hist1 + (size_t)B_ * NBINS;
    float*    p1    = (float*)(ws + OFF_P1);
    float*    p2    = (float*)(ws + OFF_P2);
    Ctl*      ctl   = (Ctl*)(ws + OFF_CTL);
    float*    out   = (float*)d_out;

    k0_zero <<<(2 * B_ * NBINS + 255) / 256, 256, 0, stream>>>(hist1);
    k1_main <<<NBLK, 256, 0, stream>>>(y_pre, y_batch, keys, hist1, p1);
    k2_sel1 <<<B_,   256, 0, stream>>>(p1, hist1, ctl);
    k3_hist2<<<NBLK, 256, 0, stream>>>(keys, ctl, hist2);
    k4_sel2 <<<B_,   256, 0, stream>>>(hist2, ctl);
    k5_sum  <<<NBLK, 256, 0, stream>>>(keys, ctl, p2);
    k6_final<<<1,    256, 0, stream>>>(p2, ctl, out);
}